// NonMaximaSuppression2d_13675175870969
// MI455X (gfx1250) — compile-verified
//
#include <hip/hip_runtime.h>
#include <stdint.h>

// NMS 3x3 on (8,16,512,512) fp32. Bandwidth-bound: ~256MB round trip -> ~11us at 23.3TB/s.
// Strategy: async-copy a full-width 512x18 row band (16 output rows + replicate halo)
// into LDS per workgroup, compute 8-neighbor max from LDS with a rolling 3-row window,
// write 16B nontemporal stores.

#define NMS_H 512
#define NMS_W 512
#define NMS_TH 16                 // output rows per workgroup
#define NMS_LROWS (NMS_TH + 2)    // staged rows incl. vertical halo
#define NMS_TILES_PER_PLANE (NMS_H / NMS_TH)

typedef float f4 __attribute__((ext_vector_type(4)));
typedef int   i4 __attribute__((ext_vector_type(4)));

// Builtin signature (from hipcc diagnostic): params are int4* in AS1 (global)
// and AS3 (LDS). Route casts through integers so we never need an
// addrspacecast: global pointers are numerically identical in AS1; generic
// LDS pointers truncate to the 32-bit LDS offset in AS3 on AMDGPU.
typedef __attribute__((address_space(1))) i4 gi4_t;
typedef __attribute__((address_space(3))) i4 li4_t;

__device__ __forceinline__ gi4_t* to_global(const void* p) {
  return (gi4_t*)(uintptr_t)p;
}
__device__ __forceinline__ li4_t* to_lds(const void* p) {
  return (li4_t*)(uint32_t)(uintptr_t)p;
}

#if __has_builtin(__builtin_amdgcn_global_load_async_to_lds_b128)
#define ASYNC_LDS_COPY_B128(g, l)                                              \
  __builtin_amdgcn_global_load_async_to_lds_b128(to_global(g), to_lds(l), 0, 0)
#else
#define ASYNC_LDS_COPY_B128(g, l)                                              \
  asm volatile("global_load_async_to_lds_b128 %0, %1, off" ::"v"(to_lds(l)),   \
               "v"(to_global(g))                                               \
               : "memory")
#endif

#if __has_builtin(__builtin_amdgcn_s_wait_asynccnt)
#define WAIT_ASYNC0() __builtin_amdgcn_s_wait_asynccnt(0)
#else
#define WAIT_ASYNC0() asm volatile("s_wait_asynccnt 0" ::: "memory")
#endif

__device__ __forceinline__ float max3f(float a, float b, float c) {
  return fmaxf(fmaxf(a, b), c);
}

__global__ __launch_bounds__(256) void nms3x3_kernel(
    const float* __restrict__ x, float* __restrict__ out) {
  __shared__ float tile[NMS_LROWS][NMS_W];  // 18*512*4 = 36 KB

  const int tid   = threadIdx.x;
  const int bid   = blockIdx.x;
  const int plane = bid >> 5;               // 32 tiles per plane
  const int trow  = bid & 31;
  const int r0    = trow * NMS_TH;

  const float* pb = x + (size_t)plane * (NMS_H * NMS_W);
  float*       po = out + (size_t)plane * (NMS_H * NMS_W);

  // ---- stage 18 x 512 floats into LDS via async global->LDS b128 copies ----
  // 18*128 = 2304 float4 chunks, 256 threads -> 9 per thread.
#pragma unroll
  for (int it = 0; it < 9; ++it) {
    const int q  = it * 256 + tid;   // float4 index within the tile
    const int lr = q >> 7;           // LDS row (128 float4 per row)
    const int lc = (q & 127) << 2;   // LDS column (float index)
    int gr = r0 - 1 + lr;            // global row with replicate clamp
    gr = gr < 0 ? 0 : gr;
    gr = gr > (NMS_H - 1) ? (NMS_H - 1) : gr;
    ASYNC_LDS_COPY_B128(pb + (size_t)gr * NMS_W + lc, &tile[lr][lc]);
  }
  WAIT_ASYNC0();
  __syncthreads();

  // ---- compute: each thread -> 4 columns x 8 rows of output ----
  const int half  = tid >> 7;            // 0..1 (which 8-row half)
  const int c     = (tid & 127) << 2;    // 0..508, step 4
  const int cm1   = (c == 0) ? 0 : c - 1;                  // clamped left
  const int cp4   = (c == NMS_W - 4) ? NMS_W - 1 : c + 4;  // clamped right
  const int rbase = half * 8;            // local output row base (0 or 8)

  // rolling 3-row window of 6 columns: [cm1, c, c+1, c+2, c+3, cp4]
  float ra[6], rb[6], rc[6];
#pragma unroll
  for (int j = 0; j < 6; ++j) {
    const int col = (j == 0) ? cm1 : (j == 5) ? cp4 : (c + j - 1);
    ra[j] = tile[rbase + 0][col];      // top row of first window
    rb[j] = tile[rbase + 1][col];      // center row of first window
  }

#pragma unroll
  for (int ry = 0; ry < 8; ++ry) {
#pragma unroll
    for (int j = 0; j < 6; ++j) {
      const int col = (j == 0) ? cm1 : (j == 5) ? cp4 : (c + j - 1);
      rc[j] = tile[rbase + ry + 2][col];   // bottom row of window
    }

    f4 res;
#pragma unroll
    for (int k = 0; k < 4; ++k) {
      // output column c+k: window cols local k, k+1, k+2
      const float cl = max3f(ra[k],     rb[k],     rc[k]);     // left column
      const float cr = max3f(ra[k + 2], rb[k + 2], rc[k + 2]); // right column
      const float tb = fmaxf(ra[k + 1], rc[k + 1]);            // center, no mid
      const float m  = max3f(cl, cr, tb);                      // 8-neighbor max
      const float xv = rb[k + 1];
      res[k] = (xv > m) ? xv : 0.0f;
    }

    // 16B nontemporal store: keep L2 for the input (halo reuse across bands)
    __builtin_nontemporal_store(
        res, (f4*)(po + (size_t)(r0 + rbase + ry) * NMS_W + c));

    // slide window down one row
#pragma unroll
    for (int j = 0; j < 6; ++j) {
      ra[j] = rb[j];
      rb[j] = rc[j];
    }
  }
}

extern "C" void kernel_launch(void* const* d_in, const int* in_sizes, int n_in,
                              void* d_out, int out_size, void* d_ws,
                              size_t ws_size, hipStream_t stream) {
  (void)n_in; (void)d_ws; (void)ws_size; (void)out_size;
  const float* x  = (const float*)d_in[0];
  float*       po = (float*)d_out;
  const int planes = in_sizes[0] / (NMS_H * NMS_W);          // 8*16 = 128
  const int blocks = planes * NMS_TILES_PER_PLANE;           // 128*32 = 4096
  nms3x3_kernel<<<dim3(blocks), dim3(256), 0, stream>>>(x, po);
}